// madgwick_filter_87960930222573
// MI455X (gfx1250) — compile-verified
//
#include <hip/hip_runtime.h>
#include <math.h>

// Madgwick MARG filter: one serial recurrence over B*T = 153,600 steps.
// Pass 1 (parallel): normalize acc/mag, pre-halve gyr -> packed 48B/step
// stream in d_ws. Pass 2 (single-lane wave32): latency-bound recurrence,
// manually software-pipelined one iteration ahead so global_load latency
// overlaps the FMA chain; v_rsq/v_sqrt on the chain; global_prefetch_b8
// lookahead; nontemporal output stores.

#define MADG_GAIN 0.041f
#define MADG_DT   (1.0f / 300.0f)

typedef __attribute__((ext_vector_type(4))) float v4f;  // native vector for NT store

__device__ __forceinline__ float fast_rsqrt(float x) {
#if __has_builtin(__builtin_amdgcn_rsqf)
    return __builtin_amdgcn_rsqf(x);      // v_rsq_f32
#else
    return 1.0f / sqrtf(x);
#endif
}

__device__ __forceinline__ float fast_sqrt(float x) {
#if __has_builtin(__builtin_amdgcn_sqrtf)
    return __builtin_amdgcn_sqrtf(x);     // v_sqrt_f32
#else
    return sqrtf(x);
#endif
}

struct Quat { float w, x, y, z; };

__device__ __forceinline__ void store_quat_nt(float4* out, int t, Quat q) {
    v4f v; v.x = q.w; v.y = q.x; v.z = q.y; v.w = q.z;
    __builtin_nontemporal_store(v, (v4f*)(out + t));
}

// One Madgwick step. gh = 0.5*gyr (pre-scaled); a, m are pre-normalized.
__device__ __forceinline__ Quat madgwick_step(Quat q,
                                              float ghx, float ghy, float ghz,
                                              float ax,  float ay,  float az,
                                              float mx,  float my,  float mz) {
    const float pw = q.w, px = q.x, py = q.y, pz = q.z;

    // qdot = q (x) (0, gh)       [gh already = 0.5*gyr]
    float d0 = -px*ghx - py*ghy - pz*ghz;
    float d1 =  pw*ghx + py*ghz - pz*ghy;
    float d2 =  pw*ghy - px*ghz + pz*ghx;
    float d3 =  pw*ghz + px*ghy - py*ghx;

    // t = q (x) (0, m);  h = t (x) conj(q)   (h0 unused)
    float t0 = -px*mx - py*my - pz*mz;
    float t1 =  pw*mx + py*mz - pz*my;
    float t2 =  pw*my - px*mz + pz*mx;
    float t3 =  pw*mz + px*my - py*mx;
    float h1 = -t0*px + t1*pw - t2*pz + t3*py;
    float h2 = -t0*py + t1*pz + t2*pw - t3*px;
    float h3 = -t0*pz - t1*py + t2*px + t3*pw;
    float bx = fast_sqrt(h1*h1 + h2*h2);
    float bz = h3;

    // Objective function f (6 components)
    float f1 = 2.0f*(px*pz - pw*py) - ax;
    float f2 = 2.0f*(pw*px + py*pz) - ay;
    float f3 = 2.0f*(0.5f - px*px - py*py) - az;
    float f4 = 2.0f*bx*(0.5f - py*py - pz*pz) + 2.0f*bz*(px*pz - pw*py) - mx;
    float f5 = 2.0f*bx*(px*py - pw*pz)        + 2.0f*bz*(pw*px + py*pz) - my;
    float f6 = 2.0f*bx*(pw*py + px*pz)        + 2.0f*bz*(0.5f - px*px - py*py) - mz;

    // grad = J^T f  (explicit expansion of the 6x4 Jacobian transpose)
    float g0 = -2.0f*py*f1 + 2.0f*px*f2
             + (-2.0f*bz*py)*f4
             + (-2.0f*bx*pz + 2.0f*bz*px)*f5
             + ( 2.0f*bx*py)*f6;
    float g1 =  2.0f*pz*f1 + 2.0f*pw*f2 - 4.0f*px*f3
             + ( 2.0f*bz*pz)*f4
             + ( 2.0f*bx*py + 2.0f*bz*pw)*f5
             + ( 2.0f*bx*pz - 4.0f*bz*px)*f6;
    float g2 = -2.0f*pw*f1 + 2.0f*pz*f2 - 4.0f*py*f3
             + (-4.0f*bx*py - 2.0f*bz*pw)*f4
             + ( 2.0f*bx*px + 2.0f*bz*pz)*f5
             + ( 2.0f*bx*pw - 4.0f*bz*py)*f6;
    float g3 =  2.0f*px*f1 + 2.0f*py*f2
             + (-4.0f*bx*pz + 2.0f*bz*px)*f4
             + (-2.0f*bx*pw + 2.0f*bz*py)*f5
             + ( 2.0f*bx*px)*f6;

    float gn = fast_rsqrt(g0*g0 + g1*g1 + g2*g2 + g3*g3);
    d0 -= MADG_GAIN * (g0 * gn);
    d1 -= MADG_GAIN * (g1 * gn);
    d2 -= MADG_GAIN * (g2 * gn);
    d3 -= MADG_GAIN * (g3 * gn);

    float w = pw + d0 * MADG_DT;
    float x = px + d1 * MADG_DT;
    float y = py + d2 * MADG_DT;
    float z = pz + d3 * MADG_DT;
    float qn = fast_rsqrt(w*w + x*x + y*y + z*z);
    Quat r; r.w = w*qn; r.x = x*qn; r.y = y*qn; r.z = z*qn;
    return r;
}

// --- Pass 1: fully parallel, q-independent preprocessing ------------------
// ws record per step (3 x float4, 48B): {0.5*g, pad}{a_hat, pad}{m_hat, pad}
__global__ void madgwick_prep_kernel(const float* __restrict__ acc,
                                     const float* __restrict__ gyr,
                                     const float* __restrict__ mag,
                                     float4* __restrict__ ws, int n) {
    int i = blockIdx.x * blockDim.x + threadIdx.x;
    if (i >= n) return;
    float ax = acc[3*i+0], ay = acc[3*i+1], az = acc[3*i+2];
    float gx = gyr[3*i+0], gy = gyr[3*i+1], gz = gyr[3*i+2];
    float mx = mag[3*i+0], my = mag[3*i+1], mz = mag[3*i+2];
    float an = fast_rsqrt(ax*ax + ay*ay + az*az);
    float mn = fast_rsqrt(mx*mx + my*my + mz*mz);
    ws[3*i+0] = make_float4(0.5f*gx, 0.5f*gy, 0.5f*gz, 0.0f);
    ws[3*i+1] = make_float4(ax*an,  ay*an,  az*an,  0.0f);
    ws[3*i+2] = make_float4(mx*mn,  my*mn,  mz*mn,  0.0f);
}

// --- Pass 2: serial recurrence, software-pipelined one iteration ahead ----
// Always loads record t+1 (record n exists in ws but is never consumed), so
// the s_wait_loadcnt lands AFTER the step-t FMA chain, not before it.
__global__ void madgwick_scan_ws_kernel(const float4* __restrict__ ws,
                                        const float* __restrict__ q0,
                                        float4* __restrict__ out, int n) {
    if (threadIdx.x != 0 || blockIdx.x != 0) return;
    float w = q0[0], x = q0[1], y = q0[2], z = q0[3];
    float qn = fast_rsqrt(w*w + x*x + y*y + z*z);
    Quat q; q.w = w*qn; q.x = x*qn; q.y = y*qn; q.z = z*qn;

    // prologue: loads for step 0
    float4 g = ws[0];
    float4 a = ws[1];
    float4 m = ws[2];
    for (int t = 0; t < n; ++t) {
        // keep the 48B/step stream hot in near caches
        __builtin_prefetch((const void*)(ws + 3*(t + 8)), 0, 3);
        // issue next iteration's loads BEFORE the dependency chain
        float4 gn_ = ws[3*(t+1) + 0];
        float4 an_ = ws[3*(t+1) + 1];
        float4 mn_ = ws[3*(t+1) + 2];
        // step t: ~60-deep FMA/rsq chain overlaps the in-flight loads
        q = madgwick_step(q, g.x, g.y, g.z, a.x, a.y, a.z, m.x, m.y, m.z);
        store_quat_nt(out, t, q);
        // rotate (this is where the load wait lands)
        g = gn_; a = an_; m = mn_;
    }
}

// --- Fallback: fused serial kernel if workspace is too small --------------
__global__ void madgwick_scan_fused_kernel(const float* __restrict__ acc,
                                           const float* __restrict__ gyr,
                                           const float* __restrict__ mag,
                                           const float* __restrict__ q0,
                                           float4* __restrict__ out, int n) {
    if (threadIdx.x != 0 || blockIdx.x != 0) return;
    float w = q0[0], x = q0[1], y = q0[2], z = q0[3];
    float qn = fast_rsqrt(w*w + x*x + y*y + z*z);
    Quat q; q.w = w*qn; q.x = x*qn; q.y = y*qn; q.z = z*qn;
    for (int t = 0; t < n; ++t) {
        __builtin_prefetch((const void*)(acc + 3*(t + 8)), 0, 3);
        __builtin_prefetch((const void*)(gyr + 3*(t + 8)), 0, 3);
        __builtin_prefetch((const void*)(mag + 3*(t + 8)), 0, 3);
        float ax = acc[3*t+0], ay = acc[3*t+1], az = acc[3*t+2];
        float gx = gyr[3*t+0], gy = gyr[3*t+1], gz = gyr[3*t+2];
        float mx = mag[3*t+0], my = mag[3*t+1], mz = mag[3*t+2];
        float an = fast_rsqrt(ax*ax + ay*ay + az*az);
        float mn = fast_rsqrt(mx*mx + my*my + mz*mz);
        q = madgwick_step(q, 0.5f*gx, 0.5f*gy, 0.5f*gz,
                          ax*an, ay*an, az*an, mx*mn, my*mn, mz*mn);
        store_quat_nt(out, t, q);
    }
}

extern "C" void kernel_launch(void* const* d_in, const int* in_sizes, int n_in,
                              void* d_out, int out_size, void* d_ws, size_t ws_size,
                              hipStream_t stream) {
    const float* acc = (const float*)d_in[0];   // (B,T,3) f32
    const float* gyr = (const float*)d_in[1];   // (B,T,3) f32
    const float* mag = (const float*)d_in[2];   // (B,T,3) f32
    const float* q0  = (const float*)d_in[3];   // (4,)    f32
    float4* out = (float4*)d_out;               // (B,T,4) f32, 16B-aligned

    const int n = in_sizes[0] / 3;              // B*T total steps
    // n+1 records: the scan's last one-ahead loop load reads one record past the end
    const size_t ws_need = (size_t)(n + 1) * 3u * sizeof(float4);

    if (ws_size >= ws_need) {
        float4* ws = (float4*)d_ws;
        madgwick_prep_kernel<<<(n + 255) / 256, 256, 0, stream>>>(acc, gyr, mag, ws, n);
        madgwick_scan_ws_kernel<<<1, 32, 0, stream>>>(ws, q0, out, n);
    } else {
        madgwick_scan_fused_kernel<<<1, 32, 0, stream>>>(acc, gyr, mag, q0, out, n);
    }
}